// GCNLayer_3582002725428
// MI455X (gfx1250) — compile-verified
//
#include <hip/hip_runtime.h>

#define N_NODES 50000
#define DIM     256
#define N_EDGES 1600000

typedef __attribute__((ext_vector_type(2))) float v2f;
typedef __attribute__((ext_vector_type(8))) float v8f;

// ---------------------------------------------------------------------------
// Kernel 1: zero the accumulator + degree arrays in workspace (float4 stores).
// ---------------------------------------------------------------------------
__global__ void gcn_zero_kernel(float4* __restrict__ ws, int n_f4) {
    int i = blockIdx.x * blockDim.x + threadIdx.x;
    if (i < n_f4) {
        ws[i] = make_float4(0.f, 0.f, 0.f, 0.f);
    }
}

// ---------------------------------------------------------------------------
// Kernel 2: edge scatter. One wave32 per edge: 32 lanes x 8 floats = 256.
// Gather feature[src] (L2-resident, 51.2MB < 192MB L2) and atomically
// accumulate into accum[dst]; lane 0 counts the degree.
// ---------------------------------------------------------------------------
__global__ void __launch_bounds__(256)
gcn_scatter_kernel(const float* __restrict__ feature,
                   const int*   __restrict__ src,
                   const int*   __restrict__ dst,
                   float*       __restrict__ accum,
                   float*       __restrict__ deg) {
    const int wave = (blockIdx.x * blockDim.x + threadIdx.x) >> 5;
    const int lane = threadIdx.x & 31;
    if (wave >= N_EDGES) return;

    const int s = src[wave];
    const int d = dst[wave];

    const float4* frow = (const float4*)(feature + (size_t)s * DIM);
    float*        arow = accum + (size_t)d * DIM;

#pragma unroll
    for (int i = 0; i < 2; ++i) {
        const int f4i = lane + i * 32;        // 0..63 float4 per row
        float4 v = frow[f4i];
        const int base = f4i * 4;
        atomicAdd(arow + base + 0, v.x);
        atomicAdd(arow + base + 1, v.y);
        atomicAdd(arow + base + 2, v.z);
        atomicAdd(arow + base + 3, v.w);
    }
    if (lane == 0) {
        atomicAdd(deg + d, 1.0f);
    }
}

// ---------------------------------------------------------------------------
// Kernel 3: finalize h = (deg>0) ? accum/deg : feature, in place over accum.
// ---------------------------------------------------------------------------
__global__ void __launch_bounds__(256)
gcn_finalize_kernel(const float* __restrict__ feature,
                    const float* __restrict__ deg,
                    float*       __restrict__ accum) {
    const int i = blockIdx.x * blockDim.x + threadIdx.x;   // float4 index
    if (i >= N_NODES * (DIM / 4)) return;
    const int row = i / (DIM / 4);
    const float dg = deg[row];

    float4 a = ((const float4*)accum)[i];
    if (dg > 0.f) {
        const float inv = 1.0f / dg;
        a.x *= inv; a.y *= inv; a.z *= inv; a.w *= inv;
    } else {
        a = ((const float4*)feature)[i];
    }
    ((float4*)accum)[i] = a;
}

// ---------------------------------------------------------------------------
// Kernel 4: out = h @ W^T + b using V_WMMA_F32_16X16X4_F32.
// One wave per (16-row M-tile, 64-col N-group); each wave computes 4 adjacent
// 16x16 N-tiles reusing a single A fragment per K-step.
//
// Fragment layout (ISA 7.12.2, 32-bit A 16x4 / B 4x16):
//   lanes 0-15 : VGPR0 = K=0, VGPR1 = K=1   (row/col = lane)
//   lanes 16-31: VGPR0 = K=2, VGPR1 = K=3   (row/col = lane-16)
// C/D: VGPR r -> M = r (lanes 0-15) / M = r+8 (lanes 16-31), N = lane&15.
// ---------------------------------------------------------------------------
__global__ void __launch_bounds__(256)
gcn_gemm_kernel(const float* __restrict__ h,
                const float* __restrict__ W,      // [DIM_OUT, DIM_IN] row-major
                const float* __restrict__ bias,
                float*       __restrict__ out) {
    const int waveId = threadIdx.x >> 5;
    const int lane   = threadIdx.x & 31;
    const int tile   = blockIdx.x * 8 + waveId;          // 0 .. 12499 (+pad)
    if (tile >= (N_NODES / 16) * 4) return;              // wave-uniform exit

    const int mtile = tile >> 2;                         // 0..3124
    const int ngrp  = tile & 3;                          // 0..3
    const int mbase = mtile * 16;
    const int nbase = ngrp * 64;

    const int l15   = lane & 15;
    const int khalf = (lane >> 4) * 2;                   // 0 or 2

    const float* arow  = h + (size_t)(mbase + l15) * DIM;       // A row (M)
    const float* brow0 = W + (size_t)(nbase +  0 + l15) * DIM;  // B col (N) = W row
    const float* brow1 = W + (size_t)(nbase + 16 + l15) * DIM;
    const float* brow2 = W + (size_t)(nbase + 32 + l15) * DIM;
    const float* brow3 = W + (size_t)(nbase + 48 + l15) * DIM;

    v8f c0 = {}, c1 = {}, c2 = {}, c3 = {};

    for (int k = 0; k < DIM; k += 4) {
        const v2f a  = *(const v2f*)(arow  + k + khalf);
        const v2f b0 = *(const v2f*)(brow0 + k + khalf);
        const v2f b1 = *(const v2f*)(brow1 + k + khalf);
        const v2f b2 = *(const v2f*)(brow2 + k + khalf);
        const v2f b3 = *(const v2f*)(brow3 + k + khalf);

        c0 = __builtin_amdgcn_wmma_f32_16x16x4_f32(false, a, false, b0,
                                                   (short)0, c0, false, false);
        c1 = __builtin_amdgcn_wmma_f32_16x16x4_f32(false, a, false, b1,
                                                   (short)0, c1, false, false);
        c2 = __builtin_amdgcn_wmma_f32_16x16x4_f32(false, a, false, b2,
                                                   (short)0, c2, false, false);
        c3 = __builtin_amdgcn_wmma_f32_16x16x4_f32(false, a, false, b3,
                                                   (short)0, c3, false, false);
    }

    // Store D + bias. VGPR r: M = mbase + r + (lane>=16 ? 8 : 0), N = l15.
    const int rofs = (lane >> 4) * 8;
    const float bb0 = bias[nbase +  0 + l15];
    const float bb1 = bias[nbase + 16 + l15];
    const float bb2 = bias[nbase + 32 + l15];
    const float bb3 = bias[nbase + 48 + l15];

#pragma unroll
    for (int r = 0; r < 8; ++r) {
        float* orow = out + (size_t)(mbase + rofs + r) * DIM;
        orow[nbase +  0 + l15] = c0[r] + bb0;
        orow[nbase + 16 + l15] = c1[r] + bb1;
        orow[nbase + 32 + l15] = c2[r] + bb2;
        orow[nbase + 48 + l15] = c3[r] + bb3;
    }
}

// ---------------------------------------------------------------------------
extern "C" void kernel_launch(void* const* d_in, const int* in_sizes, int n_in,
                              void* d_out, int out_size, void* d_ws, size_t ws_size,
                              hipStream_t stream) {
    const float* feature = (const float*)d_in[0];   // [N_NODES, DIM]
    const int*   src     = (const int*)  d_in[1];   // [N_EDGES]
    const int*   dst     = (const int*)  d_in[2];   // [N_EDGES]
    const float* W       = (const float*)d_in[3];   // [DIM, DIM]
    const float* bias    = (const float*)d_in[4];   // [DIM]
    float*       out     = (float*)d_out;           // [N_NODES, DIM]

    float* accum = (float*)d_ws;                          // N_NODES*DIM floats
    float* deg   = accum + (size_t)N_NODES * DIM;         // N_NODES floats

    // 1) zero accum + deg (both counts divisible by 4)
    const int n_f4 = (N_NODES * DIM + N_NODES) / 4;
    gcn_zero_kernel<<<(n_f4 + 255) / 256, 256, 0, stream>>>((float4*)d_ws, n_f4);

    // 2) scatter: one wave per edge
    {
        const long long threads = (long long)N_EDGES * 32;
        const int blocks = (int)((threads + 255) / 256);
        gcn_scatter_kernel<<<blocks, 256, 0, stream>>>(feature, src, dst, accum, deg);
    }

    // 3) finalize h in place over accum
    {
        const int n = N_NODES * (DIM / 4);
        gcn_finalize_kernel<<<(n + 255) / 256, 256, 0, stream>>>(feature, deg, accum);
    }

    // 4) WMMA GEMM: 12500 tiles, 8 waves (tiles) per 256-thread block
    {
        const int tiles  = (N_NODES / 16) * 4;
        const int blocks = (tiles + 7) / 8;
        gcn_gemm_kernel<<<blocks, 256, 0, stream>>>(accum, W, bias, out);
    }
}